// ConvTemporalGraphical_11261404250512
// MI455X (gfx1250) — compile-verified
//
#include <hip/hip_runtime.h>
#include <hip/hip_bf16.h>

// ---------------------------------------------------------------------------
// ConvTemporalGraphical for MI455X (gfx1250, wave32, WMMA).
// Shapes: N=64, T=12, V=256, CIN=COUT=64.
//   Phase 1: x' = conv_w @ x  (per-n GEMM 64x64 @ 64x3072)         -> WMMA f32
//   Phase 2: Aw  = relu-MLP(A[:, :7]) * A[:, 7:]  (7->16->32->64)  -> VALU+LDS
//   Phase 3: out[n,c] = x'[n,c](12x256) @ Aw[n,c](256x256)         -> WMMA f32
//
// Roofline: ~2.4 GB HBM traffic (Aw 1.07 GB write + 1.07 GB re-read dominate)
// vs ~30 GFLOP fp32 -> pure bandwidth problem, floor ~105 us @ 23.3 TB/s.
// Aw (1.07 GB) >> L2 (192 MB): stream it with non-temporal hints so L2 keeps
// x' (50 MB, reused 16x by phase-3 N-tiles).
// ---------------------------------------------------------------------------

typedef __attribute__((ext_vector_type(2))) float v2f;
typedef __attribute__((ext_vector_type(8))) float v8f;

#define N_     64
#define T_     12
#define V_     256
#define CIN_   64
#define COUT_  64
#define TV_    (T_ * V_)          // 3072
#define VV_    (V_ * V_)          // 65536
#define OUT_ELEMS ((size_t)N_ * COUT_ * T_ * V_)   // 12,582,912 floats

// ---------------------------------------------------------------------------
// Phase 1: x'[n][o][p] = sum_c conv_w[o][c] * x[n][c][p] + conv_b[o]
// One wave per (n, mtile(4), ntile(192)) 16x16 output tile; K loop 64/4 = 16
// V_WMMA_F32_16X16X4_F32 steps.
// ---------------------------------------------------------------------------
__global__ __launch_bounds__(256)
void conv1x1_x_kernel(const float* __restrict__ x,
                      const float* __restrict__ w,   // [64][64]
                      const float* __restrict__ bias,
                      float* __restrict__ xp)        // [N][64][3072]
{
    const int wid  = (blockIdx.x * blockDim.x + threadIdx.x) >> 5;
    const int lane = threadIdx.x & 31;
    const int half = lane >> 4;      // 0: lanes 0-15, 1: lanes 16-31
    const int l    = lane & 15;

    const int ntile = wid % 192;
    const int tmp   = wid / 192;
    const int mtile = tmp & 3;
    const int n     = tmp >> 2;

    const float* Xn  = x + (size_t)n * CIN_ * TV_;   // [64][3072] row-major
    const int    col = ntile * 16 + l;               // N index (t*256+v)
    const int    mrow = mtile * 16 + l;              // M index (output chan)

    v8f acc = {};
#pragma unroll
    for (int kk = 0; kk < 16; ++kk) {
        const int k0 = kk * 4 + 2 * half;
        // A fragment: W[mrow][k0], W[mrow][k0+1]  (contiguous -> b64 load)
        v2f a;
        a.x = w[mrow * CIN_ + k0];
        a.y = w[mrow * CIN_ + k0 + 1];
        // B fragment: rows k0, k0+1 of Xn at column `col` (x read once -> NT)
        v2f b;
        b.x = __builtin_nontemporal_load(&Xn[(size_t)k0       * TV_ + col]);
        b.y = __builtin_nontemporal_load(&Xn[(size_t)(k0 + 1) * TV_ + col]);
        acc = __builtin_amdgcn_wmma_f32_16x16x4_f32(
                  false, a, false, b, (short)0, acc, false, false);
    }

    float* On = xp + (size_t)n * COUT_ * TV_;
#pragma unroll
    for (int r = 0; r < 8; ++r) {
        const int m = mtile * 16 + r + 8 * half;     // C/D layout: row r (+8 hi half)
        On[(size_t)m * TV_ + col] = acc[r] + bias[m];   // x' IS reused -> regular RT
    }
}

// ---------------------------------------------------------------------------
// Phase 2: per-position MLP 7->16->32->64 (ReLU each layer) then * mask.
// One thread per (n, v, w) position; weights broadcast from LDS.
// A read once (NT loads); Aw streamed out (NT stores), coalesced across lanes.
// ---------------------------------------------------------------------------
__global__ __launch_bounds__(256)
void edge_mlp_kernel(const float* __restrict__ A,    // [N][8][V][V]
                     const float* __restrict__ w1, const float* __restrict__ b1,
                     const float* __restrict__ w2, const float* __restrict__ b2,
                     const float* __restrict__ w3, const float* __restrict__ b3,
                     float* __restrict__ aw)         // [N][64][V][V]
{
    __shared__ float sw1[16 * 7];
    __shared__ float sb1[16];
    __shared__ float sw2[32 * 16];
    __shared__ float sb2[32];
    __shared__ float sw3[64 * 32];
    __shared__ float sb3[64];

    for (int i = threadIdx.x; i < 16 * 7;  i += blockDim.x) sw1[i] = w1[i];
    for (int i = threadIdx.x; i < 16;      i += blockDim.x) sb1[i] = b1[i];
    for (int i = threadIdx.x; i < 32 * 16; i += blockDim.x) sw2[i] = w2[i];
    for (int i = threadIdx.x; i < 32;      i += blockDim.x) sb2[i] = b2[i];
    for (int i = threadIdx.x; i < 64 * 32; i += blockDim.x) sw3[i] = w3[i];
    for (int i = threadIdx.x; i < 64;      i += blockDim.x) sb3[i] = b3[i];
    __syncthreads();

    const size_t idx = (size_t)blockIdx.x * blockDim.x + threadIdx.x; // N*VV threads
    const int n   = (int)(idx >> 16);        // VV_ == 65536
    const int pos = (int)(idx & (VV_ - 1));

    const float* Ab = A + (size_t)n * 8 * VV_ + pos;

    float h0[7];
#pragma unroll
    for (int c = 0; c < 7; ++c)
        h0[c] = __builtin_nontemporal_load(&Ab[(size_t)c * VV_]);
    const float mask = __builtin_nontemporal_load(&Ab[(size_t)7 * VV_]);

    float h1[16];
#pragma unroll
    for (int o = 0; o < 16; ++o) {
        float acc = sb1[o];
#pragma unroll
        for (int c = 0; c < 7; ++c) acc = fmaf(sw1[o * 7 + c], h0[c], acc);
        h1[o] = fmaxf(acc, 0.0f);
    }

    float h2[32];
#pragma unroll
    for (int o = 0; o < 32; ++o) {
        float acc = sb2[o];
#pragma unroll
        for (int c = 0; c < 16; ++c) acc = fmaf(sw2[o * 16 + c], h1[c], acc);
        h2[o] = fmaxf(acc, 0.0f);
    }

    float* Ob = aw + (size_t)n * 64 * VV_ + pos;
#pragma unroll
    for (int o = 0; o < 64; ++o) {
        float acc = sb3[o];
#pragma unroll
        for (int c = 0; c < 32; ++c) acc = fmaf(sw3[o * 32 + c], h2[c], acc);
        __builtin_nontemporal_store(fmaxf(acc, 0.0f) * mask, &Ob[(size_t)o * VV_]);
    }
}

// ---------------------------------------------------------------------------
// Phase 3: out[n,c] (12x256) = x'[n,c] (12x256) @ Aw[n,c] (256x256)
// One wave per (n, c, ntile(16)). M=12 padded to 16 with zeroed A rows
// (multiplier trick keeps EXEC all-ones around WMMA). K loop = 256/4 = 64.
// Aw read once (NT); x' served from L2 (regular); out streamed (NT store).
// ---------------------------------------------------------------------------
__global__ __launch_bounds__(256)
void graph_gemm_kernel(const float* __restrict__ xp,  // [N][64][12][256]
                       const float* __restrict__ aw,  // [N][64][256][256]
                       float* __restrict__ out)       // [N][64][12][256]
{
    const int wid  = (blockIdx.x * blockDim.x + threadIdx.x) >> 5;
    const int lane = threadIdx.x & 31;
    const int half = lane >> 4;
    const int l    = lane & 15;

    const int ntile = wid & 15;
    const int c     = (wid >> 4) & 63;
    const int n     = wid >> 10;

    const float* X = xp + ((size_t)n * COUT_ + c) * TV_;  // [12][256]
    const float* B = aw + ((size_t)n * COUT_ + c) * VV_;  // [256][256]
    const int    col  = ntile * 16 + l;
    const int    trow = (l < T_) ? l : (T_ - 1);          // clamp, stay in-bounds
    const float  amul = (l < T_) ? 1.0f : 0.0f;           // zero-pad rows 12..15

    v8f acc = {};
#pragma unroll 8
    for (int kk = 0; kk < 64; ++kk) {
        const int k0 = kk * 4 + 2 * half;
        v2f a;
        a.x = amul * X[trow * V_ + k0];
        a.y = amul * X[trow * V_ + k0 + 1];
        v2f b;
        b.x = __builtin_nontemporal_load(&B[(size_t)k0       * V_ + col]);
        b.y = __builtin_nontemporal_load(&B[(size_t)(k0 + 1) * V_ + col]);
        acc = __builtin_amdgcn_wmma_f32_16x16x4_f32(
                  false, a, false, b, (short)0, acc, false, false);
    }

    float* O = out + ((size_t)n * COUT_ + c) * TV_;
#pragma unroll
    for (int r = 0; r < 8; ++r) {
        const int t = r + 8 * half;
        if (t < T_) __builtin_nontemporal_store(acc[r], &O[t * V_ + col]);
    }
}

// ---------------------------------------------------------------------------
extern "C" void kernel_launch(void* const* d_in, const int* in_sizes, int n_in,
                              void* d_out, int out_size, void* d_ws, size_t ws_size,
                              hipStream_t stream)
{
    const float* x      = (const float*)d_in[0];
    const float* A      = (const float*)d_in[1];
    const float* conv_w = (const float*)d_in[2];
    const float* conv_b = (const float*)d_in[3];
    const float* w1     = (const float*)d_in[4];
    const float* b1     = (const float*)d_in[5];
    const float* w2     = (const float*)d_in[6];
    const float* b2     = (const float*)d_in[7];
    const float* w3     = (const float*)d_in[8];
    const float* b3     = (const float*)d_in[9];

    float* out = (float*)d_out;              // [N][64][12][256]
    float* aw  = out + OUT_ELEMS;            // [N][64][256][256] (2nd output)
    float* xp  = (float*)d_ws;               // x' scratch: 50,331,648 bytes

    // Phase 1: 64 n * 4 mtiles * 192 ntiles = 49152 waves -> 6144 blocks
    conv1x1_x_kernel<<<6144, 256, 0, stream>>>(x, conv_w, conv_b, xp);

    // Phase 2: N*V*V = 4,194,304 threads -> 16384 blocks
    edge_mlp_kernel<<<16384, 256, 0, stream>>>(A, w1, b1, w2, b2, w3, b3, aw);

    // Phase 3: 64 n * 64 c * 16 ntiles = 65536 waves -> 8192 blocks
    graph_gemm_kernel<<<8192, 256, 0, stream>>>(xp, aw, out);
}